// CompositeLoss_62603443306866
// MI455X (gfx1250) — compile-verified
//
#include <hip/hip_runtime.h>
#include <math.h>

#define NVOX  (160 * 160 * 160)      // 4,096,000 voxels per batch element
#define VN    (2 * NVOX)             // 8,192,000 total voxels (B=2)
#define NB1   2048                   // blocks for pass1 reduction
#define NB2   2048                   // blocks for clDice sums reduction
#define SMOOTHF 1e-5f
#define EPSF    1e-6f
#define FINF  __builtin_inff()

typedef float v2f __attribute__((ext_vector_type(2)));
typedef float v8f __attribute__((ext_vector_type(8)));

#if defined(__AMDGCN__) && __has_builtin(__builtin_amdgcn_wmma_f32_16x16x4_f32)
#define USE_WMMA_REDUCE 1
#endif

#if defined(__AMDGCN__) && __has_builtin(__builtin_amdgcn_global_load_async_to_lds_b32)
#define USE_ASYNC_LDS 1
typedef __attribute__((address_space(1))) int* gas_i32p;
typedef __attribute__((address_space(3))) int* las_i32p;
#endif

// ---------------------------------------------------------------------------
// Wave(32)-wide sum via V_WMMA_F32_16X16X4_F32:
//   A (16x4): lane L<16 -> A[M=L][K=0] = v, lanes 16..31 -> A[M=L-16][K=2] = v
//   B (4x16) = ones  =>  D[m][n] = v_m + v_{m+16}  (replicated across n)
//   sum(d[0..7]) at lane0 = rows 0..7, at lane16 = rows 8..15
// Requires full EXEC (called only in uniform control flow, blockDim % 32 == 0).
// ---------------------------------------------------------------------------
__device__ __forceinline__ float wave_sum(float v) {
#ifdef USE_WMMA_REDUCE
    v2f a; a[0] = v;    a[1] = 0.0f;
    v2f b; b[0] = 1.0f; b[1] = 1.0f;
    v8f c = {0.f, 0.f, 0.f, 0.f, 0.f, 0.f, 0.f, 0.f};
    v8f d = __builtin_amdgcn_wmma_f32_16x16x4_f32(
        /*neg_a=*/false, a, /*neg_b=*/false, b,
        /*c_mod=*/(short)0, c, /*reuse_a=*/false, /*reuse_b=*/false);
    float t = ((d[0] + d[1]) + (d[2] + d[3])) + ((d[4] + d[5]) + (d[6] + d[7]));
    return __shfl(t, 0, 32) + __shfl(t, 16, 32);
#else
    #pragma unroll
    for (int off = 16; off > 0; off >>= 1) v += __shfl_down(v, off, 32);
    return __shfl(v, 0, 32);
#endif
}

// Deterministic block reduction of Q per-thread accumulators -> out[0..Q-1].
template <int Q>
__device__ __forceinline__ void block_write_partials(const float* acc, float* out) {
    __shared__ float red[8][Q];
    const int lane = threadIdx.x & 31;
    const int wv   = threadIdx.x >> 5;
    for (int q = 0; q < Q; ++q) {
        float s = wave_sum(acc[q]);
        if (lane == 0) red[wv][q] = s;
    }
    __syncthreads();
    if (threadIdx.x == 0) {
        const int nw = blockDim.x >> 5;
        for (int q = 0; q < Q; ++q) {
            float s = 0.f;
            for (int w = 0; w < nw; ++w) s += red[w][q];
            out[q] = s;
        }
    }
}

// ---------------------------------------------------------------------------
// Pass 1: fused softmax / CE / dice partial sums, materialize p_v and y_v.
// acc: [0]=ce, [1..3]=intersect_c, [4..6]=pred_sum_c, [7..9]=targ_sum_c
// ---------------------------------------------------------------------------
__global__ void pass1_kernel(const float* __restrict__ logits,
                             const long long* __restrict__ targ,
                             float* __restrict__ p, float* __restrict__ y,
                             float* __restrict__ part) {
    float acc[10];
    #pragma unroll
    for (int q = 0; q < 10; ++q) acc[q] = 0.f;

    const long long stride = (long long)gridDim.x * blockDim.x;
    for (long long i = (long long)blockIdx.x * blockDim.x + threadIdx.x; i < VN; i += stride) {
        const long long b = i / NVOX;
        const long long v = i - b * NVOX;
        const float* lp = logits + (b * 3) * (long long)NVOX + v;
        const float l0 = lp[0];
        const float l1 = lp[(long long)NVOX];
        const float l2 = lp[2 * (long long)NVOX];
        const int t = (int)targ[i];

        const float m  = fmaxf(l0, fmaxf(l1, l2));
        const float e0 = expf(l0 - m), e1 = expf(l1 - m), e2 = expf(l2 - m);
        const float s  = e0 + e1 + e2;
        const float inv = 1.0f / s;
        const float p0 = e0 * inv, p1 = e1 * inv, p2 = e2 * inv;

        const float lt = (t == 0) ? l0 : ((t == 1) ? l1 : l2);
        acc[0] -= (lt - m) - logf(s);                 // CE numerator
        const float pt = (t == 0) ? p0 : ((t == 1) ? p1 : p2);
        acc[1 + t] += pt;                             // intersect
        acc[4] += p0; acc[5] += p1; acc[6] += p2;     // pred_sum
        acc[7 + t] += 1.0f;                           // targ_sum (onehot count)

        p[i] = 1.0f - p0;                             // p_v (clip is a no-op)
        y[i] = (t == 0) ? 0.0f : 1.0f;                // y_v = 1 - onehot[:,0]
    }
    block_write_partials<10>(acc, part + (size_t)blockIdx.x * 10);
}

// ---------------------------------------------------------------------------
// 10^3 halo tile fill (8^3 outputs). Out-of-volume -> padv (matches the
// reference's +/-inf reduce_window padding). Async-to-LDS when available.
// ---------------------------------------------------------------------------
__device__ __forceinline__ void lds_fill_tile(float* tile,
                                              const float* __restrict__ src,
                                              long long base,
                                              int z0, int y0, int x0, float padv) {
    for (int li = threadIdx.x; li < 1000; li += 256) {
        const int lz = li / 100;
        const int r  = li - lz * 100;
        const int ly = r / 10;
        const int lx = r - ly * 10;
        const int gz = z0 + lz - 1, gy = y0 + ly - 1, gx = x0 + lx - 1;
        const bool in = (unsigned)gz < 160u && (unsigned)gy < 160u && (unsigned)gx < 160u;
        if (in) {
            const long long gi = base + ((long long)gz * 160 + gy) * 160 + gx;
#ifdef USE_ASYNC_LDS
            __builtin_amdgcn_global_load_async_to_lds_b32(
                (gas_i32p)(src + gi),
                (las_i32p)&tile[li],
                /*offset=*/0, /*cpol=*/0);
#else
            tile[li] = src[gi];
#endif
        } else {
            tile[li] = padv;
        }
    }
#ifdef USE_ASYNC_LDS
#if __has_builtin(__builtin_amdgcn_s_wait_asynccnt)
    __builtin_amdgcn_s_wait_asynccnt(0);
#else
    asm volatile("s_wait_asynccnt 0" ::: "memory");
#endif
#endif
    __syncthreads();
}

// eroded = 1 - maxpool(1-x)  ==  minpool(x) with +inf padding
__global__ void minpool_kernel(const float* __restrict__ src, float* __restrict__ dst) {
    __shared__ float tile[1000];
    const int x0 = blockIdx.x * 8, y0 = blockIdx.y * 8;
    const int tz = blockIdx.z;
    const int b  = tz / 20;
    const int z0 = (tz - b * 20) * 8;
    const long long base = (long long)b * NVOX;
    lds_fill_tile(tile, src, base, z0, y0, x0, FINF);
    for (int oi = threadIdx.x; oi < 512; oi += 256) {
        const int oz = oi >> 6, oy = (oi >> 3) & 7, ox = oi & 7;
        float m = FINF;
        #pragma unroll
        for (int dz = 0; dz < 3; ++dz)
            #pragma unroll
            for (int dy = 0; dy < 3; ++dy)
                #pragma unroll
                for (int dx = 0; dx < 3; ++dx)
                    m = fminf(m, tile[(oz + dz) * 100 + (oy + dy) * 10 + (ox + dx)]);
        dst[base + ((long long)(z0 + oz) * 160 + (y0 + oy)) * 160 + (x0 + ox)] = m;
    }
}

// opened = maxpool(E); x = relu(x - relu(opened - E))   (in-place on x: only
// the center voxel of x is read, so there is no cross-block hazard)
__global__ void open_update_kernel(const float* __restrict__ E, float* __restrict__ x) {
    __shared__ float tile[1000];
    const int x0 = blockIdx.x * 8, y0 = blockIdx.y * 8;
    const int tz = blockIdx.z;
    const int b  = tz / 20;
    const int z0 = (tz - b * 20) * 8;
    const long long base = (long long)b * NVOX;
    lds_fill_tile(tile, E, base, z0, y0, x0, -FINF);
    for (int oi = threadIdx.x; oi < 512; oi += 256) {
        const int oz = oi >> 6, oy = (oi >> 3) & 7, ox = oi & 7;
        float mx = -FINF;
        #pragma unroll
        for (int dz = 0; dz < 3; ++dz)
            #pragma unroll
            for (int dy = 0; dy < 3; ++dy)
                #pragma unroll
                for (int dx = 0; dx < 3; ++dx)
                    mx = fmaxf(mx, tile[(oz + dz) * 100 + (oy + dy) * 10 + (ox + dx)]);
        const float ec = tile[(oz + 1) * 100 + (oy + 1) * 10 + (ox + 1)];
        const long long gi = base + ((long long)(z0 + oz) * 160 + (y0 + oy)) * 160 + (x0 + ox);
        const float contour = fmaxf(0.f, mx - ec);
        x[gi] = fmaxf(0.f, x[gi] - contour);
    }
}

// ---------------------------------------------------------------------------
// clDice sums: acc[0]=sum(p_skel), [1]=sum(p_skel*y_v), [2]=sum(y_skel),
// [3]=sum(y_skel*p_v). p_v/y_v recomputed from inputs (saves 65MB of ws).
// ---------------------------------------------------------------------------
__global__ void cl_sums_kernel(const float* __restrict__ logits,
                               const long long* __restrict__ targ,
                               const float* __restrict__ pskel,
                               const float* __restrict__ yskel,
                               float* __restrict__ part) {
    float acc[4] = {0.f, 0.f, 0.f, 0.f};
    const long long stride = (long long)gridDim.x * blockDim.x;
    for (long long i = (long long)blockIdx.x * blockDim.x + threadIdx.x; i < VN; i += stride) {
        const long long b = i / NVOX;
        const long long v = i - b * NVOX;
        const float* lp = logits + (b * 3) * (long long)NVOX + v;
        const float l0 = lp[0];
        const float l1 = lp[(long long)NVOX];
        const float l2 = lp[2 * (long long)NVOX];
        const float m  = fmaxf(l0, fmaxf(l1, l2));
        const float e0 = expf(l0 - m), e1 = expf(l1 - m), e2 = expf(l2 - m);
        const float p_v = 1.0f - e0 / (e0 + e1 + e2);
        const float y_v = (targ[i] == 0) ? 0.0f : 1.0f;
        const float ps = pskel[i], ys = yskel[i];
        acc[0] += ps;
        acc[1] += ps * y_v;
        acc[2] += ys;
        acc[3] += ys * p_v;
    }
    block_write_partials<4>(acc, part + (size_t)blockIdx.x * 4);
}

// ---------------------------------------------------------------------------
// Final: reduce the 14 partial arrays and combine into the scalar loss.
// ---------------------------------------------------------------------------
__global__ void final_kernel(const float* __restrict__ part1,
                             const float* __restrict__ part2,
                             float* __restrict__ out) {
    __shared__ float red[8];
    __shared__ float res[14];
    for (int q = 0; q < 14; ++q) {
        float s = 0.f;
        if (q < 10) {
            for (int j = threadIdx.x; j < NB1; j += blockDim.x) s += part1[(size_t)j * 10 + q];
        } else {
            const int qq = q - 10;
            for (int j = threadIdx.x; j < NB2; j += blockDim.x) s += part2[(size_t)j * 4 + qq];
        }
        const float w = wave_sum(s);
        const int lane = threadIdx.x & 31, wv = threadIdx.x >> 5;
        if (lane == 0) red[wv] = w;
        __syncthreads();
        if (threadIdx.x == 0) {
            float t = 0.f;
            for (int i = 0; i < 8; ++i) t += red[i];
            res[q] = t;
        }
        __syncthreads();
    }
    if (threadIdx.x == 0) {
        const float ce = res[0] / (float)VN;
        float dice_sum = 0.f;
        for (int c = 0; c < 3; ++c)
            dice_sum += (2.f * res[1 + c] + SMOOTHF) / (res[4 + c] + res[7 + c] + SMOOTHF);
        const float base = ce + (1.f - dice_sum / 3.f);
        const float sp = res[10], spy = res[11], sy = res[12], syp = res[13];
        const float tprec = spy / (sp + EPSF);
        const float tsens = syp / (sy + EPSF);
        const float cld = 2.f * tprec * tsens / (tprec + tsens + EPSF);
        out[0] = base + 0.5f * (1.f - cld);
    }
}

extern "C" void kernel_launch(void* const* d_in, const int* in_sizes, int n_in,
                              void* d_out, int out_size, void* d_ws, size_t ws_size,
                              hipStream_t stream) {
    (void)in_sizes; (void)n_in; (void)out_size; (void)ws_size;
    const float*     logits = (const float*)d_in[0];
    const long long* targ   = (const long long*)d_in[1];
    float* out = (float*)d_out;

    // Workspace layout: 3 volumes (p, y, erosion scratch) + partial arrays.
    float* pA    = (float*)d_ws;
    float* yA    = pA + (size_t)VN;
    float* E     = yA + (size_t)VN;
    float* part1 = E  + (size_t)VN;              // NB1 * 10 floats
    float* part2 = part1 + (size_t)NB1 * 10;     // NB2 * 4 floats

    pass1_kernel<<<NB1, 256, 0, stream>>>(logits, targ, pA, yA, part1);

    const dim3 pg(20, 20, 40);                   // 20x20 tiles, z = b*20 + ztile
    for (int it = 0; it < 8; ++it) {
        minpool_kernel<<<pg, 256, 0, stream>>>(pA, E);
        open_update_kernel<<<pg, 256, 0, stream>>>(E, pA);
        minpool_kernel<<<pg, 256, 0, stream>>>(yA, E);
        open_update_kernel<<<pg, 256, 0, stream>>>(E, yA);
    }

    cl_sums_kernel<<<NB2, 256, 0, stream>>>(logits, targ, pA, yA, part2);
    final_kernel<<<1, 256, 0, stream>>>(part1, part2, out);
}